// EvidenceSelectionModel_87368224735664
// MI455X (gfx1250) — compile-verified
//
#include <hip/hip_runtime.h>

// Problem constants (from reference): BATCH=16, SEQ=4096, HIDDEN=1024, NUM_SENTS=32
#define BATCH 16
#define SEQ   4096
#define HID   1024
#define NS    32

typedef __attribute__((ext_vector_type(2))) float v2f;
typedef __attribute__((ext_vector_type(8))) float v8f;

// ---------------------------------------------------------------------------
// Kernel 1: per-batch sentence counts via LDS histogram -> d_ws (float[16*32])
// ---------------------------------------------------------------------------
__global__ __launch_bounds__(256) void seg_counts(const int* __restrict__ mask,
                                                  float* __restrict__ counts) {
    __shared__ unsigned int hist[NS];
    const int b = blockIdx.x;
    if (threadIdx.x < NS) hist[threadIdx.x] = 0u;
    __syncthreads();
    for (int i = threadIdx.x; i < SEQ; i += blockDim.x) {
        const int s = mask[b * SEQ + i];
        if (s >= 0 && s < NS) atomicAdd(&hist[s], 1u);
    }
    __syncthreads();
    if (threadIdx.x < NS) counts[b * NS + threadIdx.x] = (float)hist[threadIdx.x];
}

// ---------------------------------------------------------------------------
// Kernel 2: segment-sum GEMM using exact-f32 WMMA (V_WMMA_F32_16X16X4_F32).
// Grid: (HID/64 n-blocks, BATCH). Block: 128 threads = 4 waves.
// Wave w owns output tile rows 0..31 (sentences), cols n0 = blk*64 + w*16.
// A tile (16x4 f32, one-hot mask) per-lane layout:
//   lane half = lane>>4, col = lane&15
//   v0 = onehot(mask[k + 2*half + 0]), v1 = onehot(mask[k + 2*half + 1])
// B tile (4x16 f32): v0 = H[k+2*half+0][n0+col], v1 = H[k+2*half+1][n0+col]
// C/D (16x16 f32): VGPR r, lane -> (M = r + 8*half, N = n0+col)
// Division by counts folded into the (single-writer, deterministic) store.
// ---------------------------------------------------------------------------
__global__ __launch_bounds__(128) void seg_gemm(const float* __restrict__ Hs,
                                                const int* __restrict__ mask,
                                                const float* __restrict__ counts,
                                                float* __restrict__ out) {
    __shared__ int smask[SEQ];          // 16 KB of the 320 KB WGP LDS
    const int b    = blockIdx.y;
    const int wave = threadIdx.x >> 5;
    const int lane = threadIdx.x & 31;
    const int n0   = blockIdx.x * 64 + wave * 16;

    for (int i = threadIdx.x; i < SEQ; i += blockDim.x)
        smask[i] = mask[b * SEQ + i];
    __syncthreads();

    const int half = lane >> 4;
    const int col  = lane & 15;
    const float* hb = Hs + ((size_t)b * SEQ) * HID + n0 + col;

    v8f acc0 = {};   // sentences 0..15
    v8f acc1 = {};   // sentences 16..31

#pragma unroll 4
    for (int k = 0; k < SEQ; k += 4) {
        const int koff = k + 2 * half;
        const int m0 = smask[koff];       // ds_load_b64 (broadcast within half)
        const int m1 = smask[koff + 1];

        v2f a0, a1, bt;
        a0.x = (m0 == col)      ? 1.0f : 0.0f;
        a0.y = (m1 == col)      ? 1.0f : 0.0f;
        a1.x = (m0 == col + 16) ? 1.0f : 0.0f;
        a1.y = (m1 == col + 16) ? 1.0f : 0.0f;
        bt.x = hb[(size_t)koff * HID];
        bt.y = hb[(size_t)(koff + 1) * HID];

        acc0 = __builtin_amdgcn_wmma_f32_16x16x4_f32(false, a0, false, bt,
                                                     (short)0, acc0, false, false);
        acc1 = __builtin_amdgcn_wmma_f32_16x16x4_f32(false, a1, false, bt,
                                                     (short)0, acc1, false, false);
    }

    float* ob = out + (size_t)b * NS * HID + n0 + col;
    const float* cb = counts + b * NS;
#pragma unroll
    for (int r = 0; r < 8; ++r) {
        const int s0 = r + 8 * half;          // sentence row in tile 0
        ob[(size_t)s0 * HID]        = acc0[r] / cb[s0];
        ob[(size_t)(s0 + 16) * HID] = acc1[r] / cb[s0 + 16];
    }
}

// ---------------------------------------------------------------------------
// Kernel 3: write unique_sents tail (second tuple output). mode 1 = int64
// bit-pattern tail (64 f32 slots), mode 0 = one float per id.
// ---------------------------------------------------------------------------
__global__ void write_tail(float* __restrict__ out, int base, int mode) {
    const int i = threadIdx.x;
    if (i < NS) {
        if (mode) ((long long*)(out + base))[i] = (long long)i;
        else      out[base + i] = (float)i;
    }
}

extern "C" void kernel_launch(void* const* d_in, const int* in_sizes, int n_in,
                              void* d_out, int out_size, void* d_ws, size_t ws_size,
                              hipStream_t stream) {
    (void)in_sizes; (void)n_in; (void)ws_size;
    const float* Hs   = (const float*)d_in[0];   // (16, 4096, 1024) f32
    const int*   mask = (const int*)d_in[1];     // (16, 4096) int
    float* out    = (float*)d_out;
    float* counts = (float*)d_ws;                // 16*32 floats of scratch

    const int sums_n = BATCH * NS * HID;         // 524288

    seg_counts<<<BATCH, 256, 0, stream>>>(mask, counts);

    dim3 grid(HID / 64, BATCH);                  // (16, 16) workgroups, 4 waves each
    seg_gemm<<<grid, 128, 0, stream>>>(Hs, mask, counts, out);

    const int tail = out_size - sums_n;
    if (tail > 0)
        write_tail<<<1, 32, 0, stream>>>(out, sums_n, (tail >= 2 * NS) ? 1 : 0);
}